// Encoder_3693671874875
// MI455X (gfx1250) — compile-verified
//
#include <hip/hip_runtime.h>

typedef __attribute__((ext_vector_type(2))) float v2f;
typedef __attribute__((ext_vector_type(8))) float v8f;

#define F_IN  128
#define H_DIM 64

// ---------------------------------------------------------------- utilities
__global__ void zero_f32(float* __restrict__ p, long n) {
  long i = (long)blockIdx.x * blockDim.x + threadIdx.x;
  long stride = (long)gridDim.x * blockDim.x;
  for (; i < n; i += stride) p[i] = 0.0f;
}

__global__ void degree_kernel(const int* __restrict__ src, const int* __restrict__ dst,
                              float* __restrict__ deg_out, float* __restrict__ deg_in, int E) {
  int i = blockIdx.x * blockDim.x + threadIdx.x;
  int stride = gridDim.x * blockDim.x;
  for (; i < E; i += stride) {
    atomicAdd(&deg_out[src[i]], 1.0f);
    atomicAdd(&deg_in[dst[i]], 1.0f);
  }
}

// in-place deg -> deg^-0.5 with clip(deg,1)
__global__ void norm_kernel(float* __restrict__ p, int n) {
  int i = blockIdx.x * blockDim.x + threadIdx.x;
  if (i < n) p[i] = rsqrtf(fmaxf(p[i], 1.0f));
}

// Swizzle helper: stage W[K x 64] row-major into LDS in WMMA B-fragment order.
// Layout: frag[(kblock*4 + col_tile)*32 + lane] = { W[kb*4+kg][n], W[kb*4+kg+1][n] }
// with kg = 2*(lane>>4), n = col_tile*16 + (lane&15).  Inner loop then reads one
// contiguous v2f per fragment: a single ds_load_b64, zero shuffle movs.
__device__ __forceinline__ void stage_w_frag(v2f* lW, const float* __restrict__ W,
                                             int kblocks, int tid, int nthreads) {
  const int total = kblocks * 4 * 32;
  for (int i = tid; i < total; i += nthreads) {
    const int kb = i >> 7;          // / (4*32)
    const int ct = (i >> 5) & 3;
    const int ln = i & 31;
    const int kg = (ln >> 4) * 2;
    const int n  = ct * 16 + (ln & 15);
    v2f p;
    p.x = W[(kb * 4 + kg)     * H_DIM + n];
    p.y = W[(kb * 4 + kg + 1) * H_DIM + n];
    lW[i] = p;
  }
}

// -------------------------------------------------- layer-1 GEMM (128 -> 64)
// h1[N,64] = (feat * norm_src[:,None]) @ W1   via V_WMMA_F32_16X16X4_F32
__global__ __launch_bounds__(256) void gemm1_wmma(
    const float* __restrict__ feat, const float* __restrict__ norm_src,
    const float* __restrict__ W1, float* __restrict__ h1, int N) {
  __shared__ v2f lW[(F_IN / 4) * 4 * 32];  // 32 KB, fragment-order W1
  stage_w_frag(lW, W1, F_IN / 4, threadIdx.x, 256);
  __syncthreads();

  const int wave = threadIdx.x >> 5;
  const int lane = threadIdx.x & 31;
  const int tile = blockIdx.x * 8 + wave;   // 16-row output tile per wave
  const int row0 = tile * 16;
  if (row0 >= N) return;                    // wave-uniform: EXEC stays all-1s

  const int mrow = row0 + (lane & 15);
  const int kgrp = (lane >> 4) * 2;         // lanes 0-15: K {0,1}; 16-31: K {2,3}
  const float ns = norm_src[mrow];

  v8f acc[4] = {};                          // 16 x 64 output
  for (int kb = 0; kb < F_IN / 4; ++kb) {
    v2f a = *(const v2f*)(feat + (long)mrow * F_IN + kb * 4 + kgrp);
    a.x *= ns; a.y *= ns;
#pragma unroll
    for (int ct = 0; ct < 4; ++ct) {
      const v2f b = lW[(kb * 4 + ct) * 32 + lane];  // one ds_load_b64
      acc[ct] = __builtin_amdgcn_wmma_f32_16x16x4_f32(
          false, a, false, b, (short)0, acc[ct], false, false);
    }
  }

  const int mbase = row0 + ((lane >> 4) << 3);
#pragma unroll
  for (int ct = 0; ct < 4; ++ct) {
    const int n = ct * 16 + (lane & 15);
#pragma unroll
    for (int v = 0; v < 8; ++v)
      h1[(long)(mbase + v) * H_DIM + n] = acc[ct][v];
  }
}

// ------------------------------------------- edge aggregation (segment_sum)
// 16 threads per edge; float4 gather, scalar f32 atomic scatter (L2-resident)
__global__ void aggregate_kernel(const int* __restrict__ src, const int* __restrict__ dst,
                                 const float* __restrict__ h, float* __restrict__ agg, int E) {
  long t = (long)blockIdx.x * blockDim.x + threadIdx.x;
  const long total = (long)E * 16;
  const long stride = (long)gridDim.x * blockDim.x;
  for (; t < total; t += stride) {
    const int e = (int)(t >> 4);
    const int q = (int)(t & 15);
    const int s = src[e], d = dst[e];
    const float4 v = ((const float4*)h)[(long)s * 16 + q];
    float* ap = agg + (long)d * H_DIM + q * 4;
    atomicAdd(ap + 0, v.x);
    atomicAdd(ap + 1, v.y);
    atomicAdd(ap + 2, v.z);
    atomicAdd(ap + 3, v.w);
  }
}

// h2 = relu(agg1 * norm_dst + b1) * norm_src   (pre-scaled input for layer 2/3 agg)
__global__ void relu_scale_kernel(const float* __restrict__ agg, const float* __restrict__ norm_dst,
                                  const float* __restrict__ b1, const float* __restrict__ norm_src,
                                  float* __restrict__ h2, int N) {
  long i = (long)blockIdx.x * blockDim.x + threadIdx.x;
  const long total = (long)N * H_DIM;
  if (i >= total) return;
  const int node = (int)(i >> 6);
  const int j = (int)(i & 63);
  const float v = fmaxf(agg[i] * norm_dst[node] + b1[j], 0.0f);
  h2[i] = v * norm_src[node];
}

// ---------------------- mu/sigma GEMMs (shared A-fragments) + fused epilogue
// z = (agg2@W_mu * nd + b_mu) + noise * exp(agg2@W_sig * nd + b_sig)
__global__ __launch_bounds__(256) void gemm2_final_wmma(
    const float* __restrict__ agg2,
    const float* __restrict__ W_mu, const float* __restrict__ b_mu,
    const float* __restrict__ W_sig, const float* __restrict__ b_sig,
    const float* __restrict__ norm_dst, const float* __restrict__ noise,
    float* __restrict__ z, int N) {
  __shared__ v2f lWm[(H_DIM / 4) * 4 * 32];  // 16 KB each, fragment-order
  __shared__ v2f lWs[(H_DIM / 4) * 4 * 32];
  stage_w_frag(lWm, W_mu, H_DIM / 4, threadIdx.x, 256);
  stage_w_frag(lWs, W_sig, H_DIM / 4, threadIdx.x, 256);
  __syncthreads();

  const int wave = threadIdx.x >> 5;
  const int lane = threadIdx.x & 31;
  const int tile = blockIdx.x * 8 + wave;
  const int row0 = tile * 16;
  if (row0 >= N) return;

  const int mrow = row0 + (lane & 15);
  const int kgrp = (lane >> 4) * 2;

  v8f am[4] = {}, as[4] = {};
  for (int kb = 0; kb < H_DIM / 4; ++kb) {
    const v2f a = *(const v2f*)(agg2 + (long)mrow * H_DIM + kb * 4 + kgrp);
#pragma unroll
    for (int ct = 0; ct < 4; ++ct) {
      const v2f bm = lWm[(kb * 4 + ct) * 32 + lane];
      const v2f bs = lWs[(kb * 4 + ct) * 32 + lane];
      am[ct] = __builtin_amdgcn_wmma_f32_16x16x4_f32(
          false, a, false, bm, (short)0, am[ct], false, false);
      as[ct] = __builtin_amdgcn_wmma_f32_16x16x4_f32(
          false, a, false, bs, (short)0, as[ct], false, false);
    }
  }

  const int mbase = row0 + ((lane >> 4) << 3);
#pragma unroll
  for (int ct = 0; ct < 4; ++ct) {
    const int n = ct * 16 + (lane & 15);
    const float bmu = b_mu[n];
    const float bsg = b_sig[n];
#pragma unroll
    for (int v = 0; v < 8; ++v) {
      const int m = mbase + v;
      const float nd = norm_dst[m];
      const float mu = am[ct][v] * nd + bmu;
      const float ls = as[ct][v] * nd + bsg;
      z[(long)m * H_DIM + n] = mu + noise[(long)m * H_DIM + n] * expf(ls);
    }
  }
}

// ------------------------------------------------------------------ launcher
extern "C" void kernel_launch(void* const* d_in, const int* in_sizes, int n_in,
                              void* d_out, int out_size, void* d_ws, size_t ws_size,
                              hipStream_t stream) {
  const float* feat  = (const float*)d_in[0];
  const int*   src   = (const int*)d_in[1];
  const int*   dst   = (const int*)d_in[2];
  const float* noise = (const float*)d_in[3];
  const float* W1    = (const float*)d_in[4];
  const float* b1    = (const float*)d_in[5];
  const float* W_mu  = (const float*)d_in[6];
  const float* b_mu  = (const float*)d_in[7];
  const float* W_sig = (const float*)d_in[8];
  const float* b_sig = (const float*)d_in[9];

  const int E = in_sizes[1];
  const int N = in_sizes[3] / H_DIM;  // noise is [N, 64]

  float* ws       = (float*)d_ws;
  float* norm_src = ws;                          // N (deg_out then rsqrt in place)
  float* norm_dst = ws + N;                      // N
  float* buf_h    = ws + 2L * N;                 // N*64 : h1, then h2
  float* buf_agg  = buf_h + (long)N * H_DIM;     // N*64 : agg1, then agg2
  float* zout     = (float*)d_out;

  const int threads = 256;
  const int gemm_blocks = (N / 16 + 7) / 8;
  const long elems = (long)N * H_DIM;

  // degrees -> norms
  zero_f32<<<(int)((2L * N + threads - 1) / threads), threads, 0, stream>>>(norm_src, 2L * N);
  degree_kernel<<<(E + threads - 1) / threads, threads, 0, stream>>>(src, dst, norm_src, norm_dst, E);
  norm_kernel<<<(2 * N + threads - 1) / threads, threads, 0, stream>>>(norm_src, 2 * N);

  // layer 1: weight-before-aggregate (128 > 64)
  gemm1_wmma<<<gemm_blocks, threads, 0, stream>>>(feat, norm_src, W1, buf_h, N);
  zero_f32<<<(int)((elems + threads - 1) / threads), threads, 0, stream>>>(buf_agg, elems);
  aggregate_kernel<<<(int)(((long)E * 16 + threads - 1) / threads), threads, 0, stream>>>(
      src, dst, buf_h, buf_agg, E);
  relu_scale_kernel<<<(int)((elems + threads - 1) / threads), threads, 0, stream>>>(
      buf_agg, norm_dst, b1, norm_src, buf_h, N);

  // layers mu/sigma share one aggregation (weight-after-aggregate, 64 !> 64)
  zero_f32<<<(int)((elems + threads - 1) / threads), threads, 0, stream>>>(buf_agg, elems);
  aggregate_kernel<<<(int)(((long)E * 16 + threads - 1) / threads), threads, 0, stream>>>(
      src, dst, buf_h, buf_agg, E);
  gemm2_final_wmma<<<gemm_blocks, threads, 0, stream>>>(
      buf_agg, W_mu, b_mu, W_sig, b_sig, norm_dst, noise, zout, N);
}